// GCNModelVAE_71674414235792
// MI455X (gfx1250) — compile-verified
//
#include <hip/hip_runtime.h>

typedef __attribute__((ext_vector_type(2))) float v2f;
typedef __attribute__((ext_vector_type(8))) float v8f;

#define F_IN 512
#define H1   256
#define H2   64

// ---------------------------------------------------------------------------
// GEMM1: support1[M,256] = x[M,512] @ W1[512,256]
// One wave per 16x64 tile: 4 accumulators of V_WMMA_F32_16X16X4_F32.
// ---------------------------------------------------------------------------
__global__ void gemm1_wmma(const float* __restrict__ A, const float* __restrict__ B,
                           float* __restrict__ C, int M) {
    int wave = blockIdx.x * (blockDim.x >> 5) + (threadIdx.x >> 5);
    int mtiles = M >> 4;
    if (wave >= mtiles * 4) return;
    int mt = wave >> 2;          // M tile (16 rows)
    int nt = wave & 3;           // N tile of 64 cols (N=256)
    int lane = threadIdx.x & 31;
    int l15  = lane & 15;
    int half = lane >> 4;        // 0: K=0,1  1: K=2,3  (32-bit A 16x4 layout)

    const float* arow = A + (size_t)(mt * 16 + l15) * F_IN + half * 2;
    int nbase = nt * 64 + l15;

    v8f acc0 = {}, acc1 = {}, acc2 = {}, acc3 = {};
    for (int kk = 0; kk < F_IN; kk += 4) {
        v2f a = *(const v2f*)(arow + kk);
        const float* bp = B + (size_t)(kk + half * 2) * H1 + nbase;
        v2f b0 = { bp[0],  bp[H1 + 0]  };
        v2f b1 = { bp[16], bp[H1 + 16] };
        v2f b2 = { bp[32], bp[H1 + 32] };
        v2f b3 = { bp[48], bp[H1 + 48] };
        acc0 = __builtin_amdgcn_wmma_f32_16x16x4_f32(false, a, false, b0, (short)0, acc0, false, false);
        acc1 = __builtin_amdgcn_wmma_f32_16x16x4_f32(false, a, false, b1, (short)0, acc1, false, false);
        acc2 = __builtin_amdgcn_wmma_f32_16x16x4_f32(false, a, false, b2, (short)0, acc2, false, false);
        acc3 = __builtin_amdgcn_wmma_f32_16x16x4_f32(false, a, false, b3, (short)0, acc3, false, false);
    }
    // C/D layout: VGPR v -> row (v + 8*half), col = lane&15
    #pragma unroll
    for (int v = 0; v < 8; ++v) {
        float* crow = C + (size_t)(mt * 16 + half * 8 + v) * H1 + nbase;
        crow[0]  = acc0[v];
        crow[16] = acc1[v];
        crow[32] = acc2[v];
        crow[48] = acc3[v];
    }
}

// ---------------------------------------------------------------------------
// GEMM2/3 fused: support23[M,128] = relu(agg1)[M,256] @ [W2 | W3][256,64 each]
// wave nt==0 -> W2 half, nt==1 -> W3 half. ReLU fused on A load.
// ---------------------------------------------------------------------------
__global__ void gemm23_wmma(const float* __restrict__ A, const float* __restrict__ W2,
                            const float* __restrict__ W3, float* __restrict__ C, int M) {
    int wave = blockIdx.x * (blockDim.x >> 5) + (threadIdx.x >> 5);
    int mtiles = M >> 4;
    if (wave >= mtiles * 2) return;
    int mt = wave >> 1;
    int nt = wave & 1;
    int lane = threadIdx.x & 31;
    int l15  = lane & 15;
    int half = lane >> 4;

    const float* B = nt ? W3 : W2;      // [256, 64]
    const float* arow = A + (size_t)(mt * 16 + l15) * H1 + half * 2;

    v8f acc0 = {}, acc1 = {}, acc2 = {}, acc3 = {};
    for (int kk = 0; kk < H1; kk += 4) {
        v2f a = *(const v2f*)(arow + kk);
        a.x = fmaxf(a.x, 0.0f);         // relu(h1) fused
        a.y = fmaxf(a.y, 0.0f);
        const float* bp = B + (size_t)(kk + half * 2) * H2 + l15;
        v2f b0 = { bp[0],  bp[H2 + 0]  };
        v2f b1 = { bp[16], bp[H2 + 16] };
        v2f b2 = { bp[32], bp[H2 + 32] };
        v2f b3 = { bp[48], bp[H2 + 48] };
        acc0 = __builtin_amdgcn_wmma_f32_16x16x4_f32(false, a, false, b0, (short)0, acc0, false, false);
        acc1 = __builtin_amdgcn_wmma_f32_16x16x4_f32(false, a, false, b1, (short)0, acc1, false, false);
        acc2 = __builtin_amdgcn_wmma_f32_16x16x4_f32(false, a, false, b2, (short)0, acc2, false, false);
        acc3 = __builtin_amdgcn_wmma_f32_16x16x4_f32(false, a, false, b3, (short)0, acc3, false, false);
    }
    #pragma unroll
    for (int v = 0; v < 8; ++v) {
        float* crow = C + (size_t)(mt * 16 + half * 8 + v) * 128 + nt * 64 + l15;
        crow[0]  = acc0[v];
        crow[16] = acc1[v];
        crow[32] = acc2[v];
        crow[48] = acc3[v];
    }
}

// ---------------------------------------------------------------------------
// Native CDNA5 no-return float atomic: resolved in L2 (agg buffers fit in the
// 192MB L2), tracked via STOREcnt; implicit s_wait_idle at s_endpgm covers it.
// ---------------------------------------------------------------------------
__device__ __forceinline__ void atomic_fadd_b32(float* p, float v0, float v1,
                                                float v2, float v3) {
    asm volatile("global_atomic_add_f32 %0, %1, off"           :: "v"(p), "v"(v0) : "memory");
    asm volatile("global_atomic_add_f32 %0, %1, off offset:4"  :: "v"(p), "v"(v1) : "memory");
    asm volatile("global_atomic_add_f32 %0, %1, off offset:8"  :: "v"(p), "v"(v2) : "memory");
    asm volatile("global_atomic_add_f32 %0, %1, off offset:12" :: "v"(p), "v"(v3) : "memory");
}

// ---------------------------------------------------------------------------
// SpMM scatter: agg[dst] += w * feat[src], one thread per (edge, float4 chunk).
// ---------------------------------------------------------------------------
template <int H, int SH>   // H features per row, SH = log2(H/4)
__global__ void spmm_scatter(const int* __restrict__ esrc, const int* __restrict__ edst,
                             const float* __restrict__ ew, const float* __restrict__ feat,
                             float* __restrict__ agg, int E) {
    long long idx = (long long)blockIdx.x * blockDim.x + threadIdx.x;
    long long total = (long long)E << SH;
    if (idx >= total) return;
    int e = (int)(idx >> SH);
    int f = (int)(idx & ((H / 4) - 1));
    int s = esrc[e];
    int d = edst[e];
    float w = ew[e];
    float4 v = ((const float4*)(feat + (size_t)s * H))[f];
    float* p = agg + (size_t)d * H + f * 4;
    atomic_fadd_b32(p, v.x * w, v.y * w, v.z * w, v.w * w);
}

// ---------------------------------------------------------------------------
// Finalize: mu = relu(agg23[:, :64]); logvar = relu(agg23[:, 64:]);
//           z = eps * exp(logvar) + mu.  Output = [z | mu | logvar].
// ---------------------------------------------------------------------------
__global__ void finalize_kernel(const float* __restrict__ agg23, const float* __restrict__ eps,
                                float* __restrict__ out, int n /* = N*64 */) {
    int i = blockIdx.x * blockDim.x + threadIdx.x;
    if (i >= n) return;
    int row = i >> 6;
    int col = i & 63;
    float mu = fmaxf(agg23[(size_t)row * 128 + col], 0.0f);
    float lv = fmaxf(agg23[(size_t)row * 128 + 64 + col], 0.0f);
    out[i]         = eps[i] * __expf(lv) + mu;   // z
    out[n + i]     = mu;
    out[2 * n + i] = lv;
}

extern "C" void kernel_launch(void* const* d_in, const int* in_sizes, int n_in,
                              void* d_out, int out_size, void* d_ws, size_t ws_size,
                              hipStream_t stream) {
    const float* x    = (const float*)d_in[0];
    const float* W1   = (const float*)d_in[1];
    const float* W2   = (const float*)d_in[2];
    const float* W3   = (const float*)d_in[3];
    const float* ew   = (const float*)d_in[4];
    const float* eps  = (const float*)d_in[5];
    const int*   esrc = (const int*)d_in[6];
    const int*   edst = (const int*)d_in[7];
    float* out = (float*)d_out;

    const int N = in_sizes[0] / F_IN;   // 50000
    const int E = in_sizes[4];          // 1.6M

    // Workspace layout (floats). bufA is reused once support1 is dead:
    //   [0, N*256)        support1   -> later: support23 [0,N*128) + agg23 [N*128,N*256)
    //   [N*256, N*512)    agg1
    float* ws        = (float*)d_ws;
    float* support1  = ws;
    float* agg1      = ws + (size_t)N * H1;
    float* support23 = ws;
    float* agg23     = ws + (size_t)N * 128;

    const int TPB = 256;

    // 1) support1 = x @ W1
    {
        int waves  = (N / 16) * 4;
        int blocks = (waves * 32 + TPB - 1) / TPB;
        gemm1_wmma<<<blocks, TPB, 0, stream>>>(x, W1, support1, N);
    }

    // 2) agg1 = scatter_add(w * support1[src] -> dst)
    hipMemsetAsync(agg1, 0, (size_t)N * H1 * sizeof(float), stream);
    {
        long long total = (long long)E * (H1 / 4);
        int blocks = (int)((total + TPB - 1) / TPB);
        spmm_scatter<H1, 6><<<blocks, TPB, 0, stream>>>(esrc, edst, ew, support1, agg1, E);
    }

    // 3) support23 = relu(agg1) @ [W2 | W3]
    {
        int waves  = (N / 16) * 2;
        int blocks = (waves * 32 + TPB - 1) / TPB;
        gemm23_wmma<<<blocks, TPB, 0, stream>>>(agg1, W2, W3, support23, N);
    }

    // 4) agg23 = scatter_add(w * support23[src] -> dst)
    hipMemsetAsync(agg23, 0, (size_t)N * 128 * sizeof(float), stream);
    {
        long long total = (long long)E * (128 / 4);
        int blocks = (int)((total + TPB - 1) / TPB);
        spmm_scatter<128, 5><<<blocks, TPB, 0, stream>>>(esrc, edst, ew, support23, agg23, E);
    }

    // 5) z/mu/logvar
    {
        int n = N * H2;
        finalize_kernel<<<(n + TPB - 1) / TPB, TPB, 0, stream>>>(agg23, eps, out, n);
    }
}